// SimpleQNNModel_42159398977942
// MI455X (gfx1250) — compile-verified
//
#include <hip/hip_runtime.h>
#include <hip/hip_bf16.h>

#define NQ 12
#define NL 8
#define QDIM 4096          // 2^12 amplitudes
#define BATCH 1024
#define TPB 256            // 8 wave32 per workgroup

typedef __attribute__((ext_vector_type(2))) float v2f;
typedef __attribute__((ext_vector_type(8))) float v8f;

// Bank-decorrelating XOR swizzle on the LDS state layout (involution => bijection).
__device__ __forceinline__ int sa(int i) { return i ^ (((i >> 8) & 15) << 4); }

// Inverse Gray code (prefix-XOR): g(m) = m ^ (m>>1);  ginv(g(m)) == m.
__device__ __forceinline__ int ginv(int m) {
    m ^= m >> 1;
    m ^= m >> 2;
    m ^= m >> 4;
    m ^= m >> 8;
    return m;                               // stays within 12 bits
}

// Apply fused RX(thx);RZ(thz) to one amplitude pair.
// RX: t0 = c*s0 - i*s*s1 ; t1 = c*s1 - i*s*s0
// RZ: o0 = (cz - i sz)*t0 ; o1 = (cz + i sz)*t1
__device__ __forceinline__ void gate_pair(float2& a0, float2& a1,
                                          float cx, float sx, float cz, float sz) {
    float r0 = a0.x, m0 = a0.y, r1 = a1.x, m1 = a1.y;
    float t0r = cx * r0 + sx * m1;
    float t0i = cx * m0 - sx * r1;
    float t1r = cx * r1 + sx * m0;
    float t1i = cx * m1 - sx * r0;
    a0.x = cz * t0r + sz * t0i;
    a0.y = cz * t0i - sz * t0r;
    a1.x = cz * t1r - sz * t1i;
    a1.y = cz * t1i + sz * t1r;
}

// Apply 4 fused RX;RZ gates to a thread's 16 register amplitudes.
// Gate g (angles index g, wire = base+g in every mapping) acts on register-index
// bit position (3-g): strides 8,4,2,1.
__device__ __forceinline__ void reg_gates4(float2 v[16],
                                           const float cx[4], const float sx[4],
                                           const float cz[4], const float sz[4]) {
#pragma unroll
    for (int g = 0; g < 4; ++g) {
        const int bp = 3 - g;
        const int S  = 1 << bp;
#pragma unroll
        for (int p = 0; p < 8; ++p) {
            int i0 = ((p >> bp) << (bp + 1)) | (p & (S - 1));
            gate_pair(v[i0], v[i0 + S], cx[g], sx[g], cz[g], sz[g]);
        }
    }
}

// Load sin/cos of the 4 fused gates for wires [base..base+3] of layer params pl.
__device__ __forceinline__ void load_angles(const float* pl, int base,
                                            float cx[4], float sx[4],
                                            float cz[4], float sz[4]) {
#pragma unroll
    for (int g = 0; g < 4; ++g) {
        __sincosf(0.5f * pl[2 * (base + g)],     &sx[g], &cx[g]);
        __sincosf(0.5f * pl[2 * (base + g) + 1], &sz[g], &cz[g]);
    }
}

// One workgroup per batch element; statevector resident in 32 KB LDS.
// Per layer: 3 full-state LDS round trips, all 12 gates in registers, 3 barriers.
// Handoff invariant between layers: LDS slot sa(j) holds amp[g(j)] (g = Gray map),
// i.e. the CNOT chain is pre-applied by the producing phase's scatter, so the
// consuming Phase A reads contiguously.
__global__ __launch_bounds__(TPB) void qnn_state_kernel(const float* __restrict__ x,
                                                        const float* __restrict__ params,
                                                        float* __restrict__ zp /*BATCH x 16*/) {
    const int b   = blockIdx.x;
    const int tid = threadIdx.x;

    __shared__ float2 st[QDIM];             // interleaved (re, im): 32 KB

    // ---- Encoding layer in closed form: product state after 12 RX on |0..0> ----
    // amp(i) = [prod_w mag_w(bit)] * (-i)^popcount(i)
    float cw[NQ], sw[NQ];
#pragma unroll
    for (int w = 0; w < NQ; ++w) {
        float th = x[b * NQ + w] * (3.14159265358979323846f * 0.5f);
        __sincosf(th, &sw[w], &cw[w]);
    }
    // bits 4..11 of i come from tid (i = tid*16 + k); bit position bp -> wire 11-bp
    float himag = 1.0f;
    int   hipop = 0;
#pragma unroll
    for (int bp = 4; bp < 12; ++bp) {
        int bit = (tid >> (bp - 4)) & 1;
        himag *= bit ? sw[11 - bp] : cw[11 - bp];
        hipop += bit;
    }
    float2 v[16];
#pragma unroll
    for (int k = 0; k < 16; ++k) {
        float m = himag;
        m *= (k & 1) ? sw[11] : cw[11];
        m *= (k & 2) ? sw[10] : cw[10];
        m *= (k & 4) ? sw[9]  : cw[9];
        m *= (k & 8) ? sw[8]  : cw[8];
        int pop = (hipop + __popc(k)) & 3;  // phase (-i)^pop
        float2 a;
        a.x = (pop == 0) ? m : ((pop == 2) ? -m : 0.0f);
        a.y = (pop == 1) ? -m : ((pop == 3) ? m : 0.0f);
        v[k] = a;
    }
    // Scatter through g^{-1}: slot sa(j) <- amp[g(j)], so layer 0 reads contiguously.
#pragma unroll
    for (int k = 0; k < 16; ++k) st[sa(ginv(tid * 16 + k))] = v[k];
    __syncthreads();

    // ---- Variational layers ----
    for (int l = 0; l < NL; ++l) {
        const float* pl = params + l * (2 * NQ);
        const bool last = (l == NL - 1);
        float cx[4], sxn[4], cz[4], sz[4];

        // Phase A (mapping1: thread owns index bits 3..0).
        // Slot sa(tid*16+k) already holds post-CNOT state at index tid*16+k.
        // Contiguous read, wires 8..11 in registers, contiguous write-back to the
        // SAME slots by the SAME thread -> no internal barrier.
        load_angles(pl, 8, cx, sxn, cz, sz);
#pragma unroll
        for (int k = 0; k < 16; ++k) v[k] = st[sa(tid * 16 + k)];
        reg_gates4(v, cx, sxn, cz, sz);
#pragma unroll
        for (int k = 0; k < 16; ++k) st[sa(tid * 16 + k)] = v[k];
        __syncthreads();

        // Phase B (mapping2: thread owns index bits 7..4): wires 4..7 in registers.
        load_angles(pl, 4, cx, sxn, cz, sz);
#pragma unroll
        for (int k = 0; k < 16; ++k) {
            int i2 = ((tid >> 4) << 8) | (k << 4) | (tid & 15);
            v[k] = st[sa(i2)];
        }
        reg_gates4(v, cx, sxn, cz, sz);
#pragma unroll
        for (int k = 0; k < 16; ++k) {       // same addresses, same thread: no barrier needed
            int i2 = ((tid >> 4) << 8) | (k << 4) | (tid & 15);
            st[sa(i2)] = v[k];
        }
        __syncthreads();

        // Phase C (mapping3: thread owns index bits 11..8): wires 0..3 in registers.
        // Layers 0..6 scatter through g^{-1} (pre-applies next layer's CNOT chain);
        // the last layer stores canonically for the expectation pass.
        load_angles(pl, 0, cx, sxn, cz, sz);
#pragma unroll
        for (int k = 0; k < 16; ++k) v[k] = st[sa((k << 8) | tid)];
        reg_gates4(v, cx, sxn, cz, sz);
#pragma unroll
        for (int k = 0; k < 16; ++k) {
            int i3  = (k << 8) | tid;
            int dst = last ? i3 : ginv(i3);
            st[sa(dst)] = v[k];
        }
        __syncthreads();
    }

    // ---- Pauli-Z expectations ----
    // Wires 0..7 have thread-constant sign (bits 4..11 fixed per block of 16);
    // only wires 8..11 vary inside the block.
    float ptot = 0.0f, a8 = 0.0f, a9 = 0.0f, a10 = 0.0f, a11 = 0.0f;
#pragma unroll
    for (int k = 0; k < 16; ++k) {
        float2 a = st[sa(tid * 16 + k)];
        float p  = a.x * a.x + a.y * a.y;
        ptot += p;
        a8  += (k & 8) ? -p : p;   // wire 8  -> bit 3 of k
        a9  += (k & 4) ? -p : p;   // wire 9  -> bit 2
        a10 += (k & 2) ? -p : p;   // wire 10 -> bit 1
        a11 += (k & 1) ? -p : p;   // wire 11 -> bit 0
    }
    float acc[NQ];
#pragma unroll
    for (int w = 0; w < 8; ++w)
        acc[w] = ((tid >> (7 - w)) & 1) ? -ptot : ptot;  // wire w -> bit (7-w) of tid
    acc[8] = a8; acc[9] = a9; acc[10] = a10; acc[11] = a11;

    __syncthreads();
    float* red = (float*)st;                 // reuse state LDS as [NQ][TPB] scratch
#pragma unroll
    for (int w = 0; w < NQ; ++w) red[w * TPB + tid] = acc[w];
    __syncthreads();
    if (tid < NQ) {
        float s = 0.0f;
        for (int t = 0; t < TPB; ++t) s += red[tid * TPB + t];
        zp[b * 16 + tid] = s;
    } else if (tid < 16) {
        zp[b * 16 + tid] = 0.0f;             // K padding
    }
}

// Head: out[b] = sum_j relu( sum_k z[b][k] * W[k][j] )
// One wave per 16-row tile; K=12 via 3 chained V_WMMA_F32_16X16X4_F32.
__global__ __launch_bounds__(32) void qnn_linear_wmma(const float* __restrict__ zp,
                                                      const float* __restrict__ W,
                                                      float* __restrict__ out) {
    const int tile = blockIdx.x;        // 64 tiles of 16 batch rows
    const int lane = threadIdx.x;       // wave32
    const int m    = lane & 15;
    const int kb   = (lane >> 4) << 1;  // 0 for lanes 0-15, 2 for lanes 16-31
    const int row  = tile * 16 + m;
    const int mc   = (m < NQ) ? m : 0;  // branchless pad

    v8f acc = {};
#pragma unroll
    for (int c = 0; c < 3; ++c) {
        int k0 = 4 * c + kb;            // K index of A VGPR0 for this lane
        v2f a, bmat;
        // A 16x4 f32 layout: lanes 0-15 hold {K0,K1}, lanes 16-31 hold {K2,K3}
        a.x = zp[row * 16 + k0];
        a.y = zp[row * 16 + k0 + 1];
        // B 4x16 f32: row K striped across lanes; pad columns N>=12 with 0
        float b0 = W[k0 * NQ + mc];
        float b1 = W[(k0 + 1) * NQ + mc];
        bmat.x = (m < NQ) ? b0 : 0.0f;
        bmat.y = (m < NQ) ? b1 : 0.0f;
        acc = __builtin_amdgcn_wmma_f32_16x16x4_f32(
            /*neg_a=*/false, a, /*neg_b=*/false, bmat,
            /*c_mod=*/(short)0, acc, /*reuse_a=*/false, /*reuse_b=*/false);
    }

    // D 16x16 f32 layout: VGPR r -> M = r (lanes 0-15) / r+8 (lanes 16-31), N = lane&15
    __shared__ float dmat[16][17];
    const int mbase = (lane < 16) ? 0 : 8;
    const int n     = lane & 15;
#pragma unroll
    for (int r = 0; r < 8; ++r) {
        float vv = acc[r];
        dmat[mbase + r][n] = vv > 0.0f ? vv : 0.0f;   // ReLU
    }
    __syncthreads();
    if (lane < 16) {
        float s = 0.0f;
#pragma unroll
        for (int j = 0; j < NQ; ++j) s += dmat[lane][j];  // cols >=12 are exact zeros
        out[tile * 16 + lane] = s;
    }
}

extern "C" void kernel_launch(void* const* d_in, const int* in_sizes, int n_in,
                              void* d_out, int out_size, void* d_ws, size_t ws_size,
                              hipStream_t stream) {
    const float* x      = (const float*)d_in[0];   // (1024, 12)
    const float* params = (const float*)d_in[1];   // (8, 24)
    const float* W      = (const float*)d_in[2];   // (12, 12)
    float* out = (float*)d_out;                    // (1024,)
    float* zp  = (float*)d_ws;                     // 1024 x 16 padded z

    qnn_state_kernel<<<BATCH, TPB, 0, stream>>>(x, params, zp);
    qnn_linear_wmma<<<BATCH / 16, 32, 0, stream>>>(zp, W, out);
}